// SSIMLoss_81956565943036
// MI455X (gfx1250) — compile-verified
//
#include <hip/hip_runtime.h>

// ---------------- SSIM-3D via separable Gaussian conv on WMMA ---------------
// Each 11-tap 1D conv over a 16-output window is a 16x28 (A) * 28x16 (B)
// matmul where B is the constant Toeplitz band matrix of the Gaussian taps.
// 7 k-steps of V_WMMA_F32_16X16X4_F32 per tile per field. The Toeplitz matrix
// is built once per block in LDS (avoids per-lane select trees).

#define PAD     5
#define TDIM    16
#define KCOLS   26          // TDIM + 11 - 1
#define LDSW    28          // padded K (7 * 4) == LDS row stride
#define NKSTEP  7
#define WAVES   8
#define BLOCK   256
#define GSIZE   (LDSW * 16) // 448-entry Toeplitz table

#define DD 64
#define HH 192
#define WW 192
#define C1 1.0e-4f
#define C2 9.0e-4f

typedef float v2f __attribute__((ext_vector_type(2)));
typedef float v8f __attribute__((ext_vector_type(8)));

// Build the normalized Gaussian taps and the 28x16 Toeplitz matrix
// G[j][n] = g[j-n] for 0 <= j-n <= 10, j < 26, else 0 -- once per block.
__device__ __forceinline__ void build_gtab(float* gs, float* Gt, int tid) {
    if (tid < 11) {
        float d = (float)(tid - 5);
        gs[tid] = expf(-(d * d) / 4.5f);    // sigma = 1.5 -> 2*sigma^2 = 4.5
    }
    __syncthreads();
    float sum = gs[0] + gs[1] + gs[2] + gs[3] + gs[4] + gs[5] +
                gs[6] + gs[7] + gs[8] + gs[9] + gs[10];
    float inv = 1.f / sum;
#pragma unroll
    for (int idx = tid; idx < GSIZE; idx += BLOCK) {   // 2 iterations
        int j = idx >> 4, n = idx & 15;
        int d = j - n;
        float v = 0.f;
        if (d >= 0 && d <= 10 && j < KCOLS) v = gs[d] * inv;  // LDS gather, no selects
        Gt[idx] = v;
    }
    __syncthreads();
}

// B fragment layout mirrors f32 A 16x4: lane&15 -> N, K = 4*kk + 2*(lane>=16) + vgpr.
__device__ __forceinline__ void load_bfrags(const float* Gt, int lane, v2f* bf) {
    int n = lane & 15, hi = lane >> 4;
#pragma unroll
    for (int kk = 0; kk < NKSTEP; ++kk) {
        int j = 4 * kk + 2 * hi;
        bf[kk][0] = Gt[j * 16 + n];
        bf[kk][1] = Gt[(j + 1) * 16 + n];
    }
}

// Gather a 16-row x 26-col tile into LDS (row stride LDSW), zero padded along
// the conv axis. Address = base + r*rowStride + a*axisStride, a = a0 + c - 5.
template <bool COL_FAST>
__device__ __forceinline__ void load_tile(const float* __restrict__ src, float* lds,
                                          int base, int rowStride, int axisStride,
                                          int a0, int axisLen, int lane) {
#pragma unroll
    for (int i = 0; i < 13; ++i) {          // 13*32 == 16*26
        int idx = lane + 32 * i;
        int r, c;
        if (COL_FAST) { r = idx / KCOLS; c = idx - KCOLS * r; }  // axis contiguous
        else          { r = idx & 15;   c = idx >> 4; }          // rows contiguous
        int a = a0 + c - PAD;
        float v = 0.f;
        if (a >= 0 && a < axisLen) v = src[base + r * rowStride + a * axisStride];
        lds[r * LDSW + c] = v;
    }
    if (lane < 16) { lds[lane * LDSW + 26] = 0.f; lds[lane * LDSW + 27] = 0.f; }
}

// 7 k-steps of f32 WMMA: out[m][n] = sum_k A[m][k] * B[k][n]
__device__ __forceinline__ v8f conv_wmma(const float* lds, const v2f* bf, int lane) {
    v8f acc = {0.f, 0.f, 0.f, 0.f, 0.f, 0.f, 0.f, 0.f};
    int m = lane & 15, hi = lane >> 4;
#pragma unroll
    for (int kk = 0; kk < NKSTEP; ++kk) {
        int k = 4 * kk + 2 * hi;
        v2f a;
        a[0] = lds[m * LDSW + k];
        a[1] = lds[m * LDSW + k + 1];
        acc = __builtin_amdgcn_wmma_f32_16x16x4_f32(false, a, false, bf[kk],
                                                    (short)0, acc, false, false);
    }
    return acc;
}

// -------- Stage 1: conv along W for 5 fields (p, t, p^2, t^2, p*t) ----------
__global__ void __launch_bounds__(BLOCK)
ssim_stage1_wconv(const float* __restrict__ pred, const float* __restrict__ targ,
                  float* __restrict__ outA, int slabB0, int Bs) {
    __shared__ float gs[11];
    __shared__ float Gt[GSIZE];
    __shared__ float ldsP[WAVES][TDIM * LDSW];
    __shared__ float ldsT[WAVES][TDIM * LDSW];
    int tid = threadIdx.x, lane = tid & 31, wv = tid >> 5;
    build_gtab(gs, Gt, tid);
    v2f bf[NKSTEP]; load_bfrags(Gt, lane, bf);

    int t = blockIdx.x * WAVES + wv;
    int wt = t % 12; t /= 12;
    int ht = t % 12; t /= 12;
    int d = t % DD;  int bb = t / DD;
    int b = slabB0 + bb;
    int h0 = ht * 16, w0 = wt * 16;

    int baseIn = ((b * DD + d) * HH + h0) * WW;  // rows = h, axis = w
    load_tile<true>(pred, ldsP[wv], baseIn, WW, 1, w0, WW, lane);
    load_tile<true>(targ, ldsT[wv], baseIn, WW, 1, w0, WW, lane);
    __syncthreads();

    int m = lane & 15, hi = lane >> 4, n = lane & 15;
    v8f aP = {0,0,0,0,0,0,0,0}, aT = aP, aPP = aP, aTT = aP, aPT = aP;
#pragma unroll
    for (int kk = 0; kk < NKSTEP; ++kk) {
        int k = 4 * kk + 2 * hi;
        v2f p, q;
        p[0] = ldsP[wv][m * LDSW + k];  p[1] = ldsP[wv][m * LDSW + k + 1];
        q[0] = ldsT[wv][m * LDSW + k];  q[1] = ldsT[wv][m * LDSW + k + 1];
        v2f pp = p * p, qq = q * q, pq = p * q;
        aP  = __builtin_amdgcn_wmma_f32_16x16x4_f32(false, p,  false, bf[kk], (short)0, aP,  false, false);
        aT  = __builtin_amdgcn_wmma_f32_16x16x4_f32(false, q,  false, bf[kk], (short)0, aT,  false, false);
        aPP = __builtin_amdgcn_wmma_f32_16x16x4_f32(false, pp, false, bf[kk], (short)0, aPP, false, false);
        aTT = __builtin_amdgcn_wmma_f32_16x16x4_f32(false, qq, false, bf[kk], (short)0, aTT, false, false);
        aPT = __builtin_amdgcn_wmma_f32_16x16x4_f32(false, pq, false, bf[kk], (short)0, aPT, false, false);
    }

    int slabElems = Bs * DD * HH * WW;
    int outBase = ((bb * DD + d) * HH) * WW;
#pragma unroll
    for (int r = 0; r < 8; ++r) {
        int M = r + 8 * hi;                         // h offset
        int addr = outBase + (h0 + M) * WW + (w0 + n);
        outA[0 * slabElems + addr] = aP[r];
        outA[1 * slabElems + addr] = aT[r];
        outA[2 * slabElems + addr] = aPP[r];
        outA[3 * slabElems + addr] = aTT[r];
        outA[4 * slabElems + addr] = aPT[r];
    }
}

// -------- Stage 2: conv along H (rows = contiguous w), LDS-transposed store -
__global__ void __launch_bounds__(BLOCK)
ssim_stage2_hconv(const float* __restrict__ inA, float* __restrict__ outB, int Bs) {
    __shared__ float gs[11];
    __shared__ float Gt[GSIZE];
    __shared__ float lds[WAVES][TDIM * LDSW];
    __shared__ float ldsO[WAVES][TDIM * 17];
    int tid = threadIdx.x, lane = tid & 31, wv = tid >> 5;
    build_gtab(gs, Gt, tid);
    v2f bf[NKSTEP]; load_bfrags(Gt, lane, bf);

    int t = blockIdx.x * WAVES + wv;
    int ht = t % 12; t /= 12;
    int wt = t % 12; t /= 12;
    int d = t % DD;  int bb = t / DD;
    int h0 = ht * 16, w0 = wt * 16;
    int slabElems = Bs * DD * HH * WW;
    int n = lane & 15, hi = lane >> 4;

    for (int f = 0; f < 5; ++f) {
        int base = f * slabElems + ((bb * DD + d) * HH) * WW + w0;  // rows=w, axis=h
        load_tile<false>(inA, lds[wv], base, 1, WW, h0, HH, lane);
        __syncthreads();
        v8f acc = conv_wmma(lds[wv], bf, lane);
        // transpose via LDS so stores run along contiguous w
#pragma unroll
        for (int r = 0; r < 8; ++r) ldsO[wv][(r + 8 * hi) * 17 + n] = acc[r];
        __syncthreads();
#pragma unroll
        for (int j = 0; j < 8; ++j) {
            int flat = lane + 32 * j;               // 0..255
            int w_off = flat & 15, h_off = flat >> 4;
            float v = ldsO[wv][w_off * 17 + h_off];
            outB[f * slabElems + ((bb * DD + d) * HH + h0 + h_off) * WW + w0 + w_off] = v;
        }
        __syncthreads();
    }
}

// -------- Stage 3: conv along D + SSIM map + per-tile partial sum -----------
__global__ void __launch_bounds__(BLOCK)
ssim_stage3_dconv(const float* __restrict__ inB, float* __restrict__ partials,
                  int slabB0, int Bs) {
    __shared__ float gs[11];
    __shared__ float Gt[GSIZE];
    __shared__ float lds[WAVES][TDIM * LDSW];
    int tid = threadIdx.x, lane = tid & 31, wv = tid >> 5;
    build_gtab(gs, Gt, tid);
    v2f bf[NKSTEP]; load_bfrags(Gt, lane, bf);

    int t = blockIdx.x * WAVES + wv;
    int dt = t % 4;   t /= 4;
    int wt = t % 12;  t /= 12;
    int h  = t % HH;  int bb = t / HH;
    int b = slabB0 + bb;
    int d0 = dt * 16, w0 = wt * 16;
    int slabElems = Bs * DD * HH * WW;

    v8f acc[5];
#pragma unroll
    for (int f = 0; f < 5; ++f) {
        int base = f * slabElems + (bb * DD * HH + h) * WW + w0;   // rows=w, axis=d
        load_tile<false>(inB, lds[wv], base, 1, HH * WW, d0, DD, lane);
        __syncthreads();
        acc[f] = conv_wmma(lds[wv], bf, lane);
        __syncthreads();
    }

    float s = 0.f;
#pragma unroll
    for (int r = 0; r < 8; ++r) {
        float mp = acc[0][r], mt = acc[1][r];
        float ep2 = acc[2][r], et2 = acc[3][r], ept = acc[4][r];
        float mp2 = mp * mp, mt2 = mt * mt, mpt = mp * mt;
        float sp2 = ep2 - mp2, st2 = et2 - mt2, spt = ept - mpt;
        float num = (2.f * mpt + C1) * (2.f * spt + C2);
        float den = (mp2 + mt2 + C1) * (sp2 + st2 + C2);
        s += num / den;
    }
#pragma unroll
    for (int off = 16; off > 0; off >>= 1) s += __shfl_xor(s, off, 32);
    if (lane == 0) {
        int slot = ((b * HH + h) * 12 + wt) * 4 + dt;   // unique per tile
        partials[slot] = s;
    }
}

// -------- Final deterministic reduction -------------------------------------
__global__ void ssim_finalize(const float* __restrict__ partials, float* __restrict__ out,
                              int n, float invN) {
    __shared__ float red[256];
    float s = 0.f;
    for (int i = threadIdx.x; i < n; i += 256) s += partials[i];
    red[threadIdx.x] = s;
    __syncthreads();
    for (int st = 128; st > 0; st >>= 1) {
        if ((int)threadIdx.x < st) red[threadIdx.x] += red[threadIdx.x + st];
        __syncthreads();
    }
    if (threadIdx.x == 0) out[0] = 1.f - red[0] * invN;
}

extern "C" void kernel_launch(void* const* d_in, const int* in_sizes, int n_in,
                              void* d_out, int out_size, void* d_ws, size_t ws_size,
                              hipStream_t stream) {
    (void)in_sizes; (void)n_in; (void)out_size;
    const float* pred = (const float*)d_in[0];
    const float* targ = (const float*)d_in[1];
    float* out = (float*)d_out;

    const int B = 4;
    const int NPART = B * HH * 12 * 4;                 // 36864 tiles total
    const size_t partBytes = (size_t)NPART * sizeof(float);   // 144 KB, 256B aligned

    // Pick the largest batch slab (4,2,1) whose two 5-field buffers fit in ws.
    int Bs = 4;
    while (Bs > 1 && partBytes + 10ull * (size_t)Bs * DD * HH * WW * 4ull > ws_size)
        Bs >>= 1;

    float* partials = (float*)d_ws;
    size_t slabElems = (size_t)Bs * DD * HH * WW;
    float* bufA = (float*)((char*)d_ws + partBytes);
    float* bufB = bufA + 5 * slabElems;

    int tilesPerSlab = Bs * DD * 12 * 12;              // == Bs * 9216 (all stages)
    dim3 blk(BLOCK);
    dim3 grd(tilesPerSlab / WAVES);

    for (int s0 = 0; s0 < B; s0 += Bs) {
        ssim_stage1_wconv<<<grd, blk, 0, stream>>>(pred, targ, bufA, s0, Bs);
        ssim_stage2_hconv<<<grd, blk, 0, stream>>>(bufA, bufB, Bs);
        ssim_stage3_dconv<<<grd, blk, 0, stream>>>(bufB, partials, s0, Bs);
    }
    ssim_finalize<<<1, 256, 0, stream>>>(partials, out, NPART,
                                         1.f / (float)(B * DD * HH * WW));
}